// BiFormer_78881369359061
// MI455X (gfx1250) — compile-verified
//
#include <hip/hip_runtime.h>
#include <math.h>

// ---------------- problem constants ----------------
#define B_    4
#define N_    4096
#define F_    15
#define A_    24
#define AFN_  14
#define HD_   10
#define KTOP  204
#define QK_STR 12              // q/k/v row stride (10 + 2 zero pad for WMMA K=12)
#define SSTR  4100             // LDS score row stride (4096 + 4 pad: breaks 2-way bank conflict)
#define SCALE_ 0.3162277660168379f   // 1/sqrt(10)

typedef __attribute__((ext_vector_type(2))) float v2f;
typedef __attribute__((ext_vector_type(8))) float v8f;

// =====================================================================
// Stage 1: per-sequence conv features (6 convs, relu, adaptive max pool)
// fused with the Q/K/V projections. One wave per sequence.
// =====================================================================
__global__ __launch_bounds__(128)
void feat_qkv_kernel(const float* __restrict__ x,
    const float* __restrict__ cw2, const float* __restrict__ cb2,
    const float* __restrict__ cw3, const float* __restrict__ cb3,
    const float* __restrict__ cw4, const float* __restrict__ cb4,
    const float* __restrict__ cw5, const float* __restrict__ cb5,
    const float* __restrict__ cw6, const float* __restrict__ cb6,
    const float* __restrict__ cw7, const float* __restrict__ cb7,
    const float* __restrict__ w_q, const float* __restrict__ b_q,
    const float* __restrict__ w_k, const float* __restrict__ b_k,
    const float* __restrict__ w_v, const float* __restrict__ b_v,
    float* __restrict__ qb, float* __restrict__ kb, float* __restrict__ vb)
{
    __shared__ float xs[4][F_ * A_];   // 4 sequences per block
    __shared__ float wt[840];          // all conv weights, packed
    __shared__ float bs[AFN_];
    __shared__ float ft[4][AFN_];

    const int tid  = threadIdx.x;
    const int wv   = tid >> 5;
    const int lane = tid & 31;

    // stage conv weights into LDS (packed: h2|h3|h4|h5|h6|h7)
    for (int i = tid; i <  90; i += 128) wt[      i] = cw2[i];
    for (int i = tid; i < 135; i += 128) wt[ 90 + i] = cw3[i];
    for (int i = tid; i < 180; i += 128) wt[225 + i] = cw4[i];
    for (int i = tid; i < 150; i += 128) wt[405 + i] = cw5[i];
    for (int i = tid; i < 180; i += 128) wt[555 + i] = cw6[i];
    for (int i = tid; i < 105; i += 128) wt[735 + i] = cw7[i];
    if (tid < AFN_) {
        float v;
        if      (tid <  3) v = cb2[tid];
        else if (tid <  6) v = cb3[tid - 3];
        else if (tid <  9) v = cb4[tid - 6];
        else if (tid < 11) v = cb5[tid - 9];
        else if (tid < 13) v = cb6[tid - 11];
        else               v = cb7[0];
        bs[tid] = v;
    }

    const int seq = blockIdx.x * 4 + wv;
    const float* xrow = x + (size_t)seq * (F_ * A_);
    for (int i = lane; i < F_ * A_; i += 32) xs[wv][i] = xrow[i];
    __syncthreads();

    // lanes 0..13 each own one output filter
    if (lane < AFN_) {
        const int hTab[AFN_] = {2,2,2, 3,3,3, 4,4,4, 5,5, 6,6, 7};
        const int oTab[AFN_] = {0,30,60, 90,135,180, 225,285,345, 405,480, 555,645, 735};
        const int hh = hTab[lane];
        const float* w = &wt[oTab[lane]];
        const float bias = bs[lane];
        float best = 0.0f;                       // relu output is >= 0
        for (int p = 0; p <= A_ - hh; ++p) {
            float s = bias;
            for (int c = 0; c < F_; ++c) {
                const float* xr = &xs[wv][c * A_ + p];
                const float* wr = &w[c * hh];
                for (int kk = 0; kk < hh; ++kk) s = fmaf(xr[kk], wr[kk], s);
            }
            best = fmaxf(best, fmaxf(s, 0.0f));  // relu then maxpool
        }
        ft[wv][lane] = best;
    }
    __syncthreads();

    // lanes 0..29: q(0-9) / k(10-19) / v(20-29); lanes 30,31 write zero pads
    if (lane < 30) {
        const int which = lane / 10, d = lane % 10;
        const float* W  = which == 0 ? w_q : (which == 1 ? w_k : w_v);
        const float* Bv = which == 0 ? b_q : (which == 1 ? b_k : b_v);
        float acc = Bv[d];
        for (int i = 0; i < AFN_; ++i) acc = fmaf(W[d * AFN_ + i], ft[wv][i], acc);
        float* dst = which == 0 ? qb : (which == 1 ? kb : vb);
        dst[(size_t)seq * QK_STR + d] = acc;
    } else {
        for (int j = lane - 30; j < 6; j += 2) {
            float* dst = j < 2 ? qb : (j < 4 ? kb : vb);
            dst[(size_t)seq * QK_STR + 10 + (j & 1)] = 0.0f;
        }
    }
}

// =====================================================================
// Stage 2: one workgroup per (batch, 16-query tile).
//   - WMMA (f32 16x16x4, K padded to 12) the full 16x4096 score stripe
//     into LDS (16*4100*4 = 262 KB of the 320 KB WGP LDS)
//   - per-row top-204 (one wave per row, lane-cached strip maxima)
//   - softmax, probs out, gathered probs @ V, output projections
//   - deterministic per-workgroup partial logits to workspace
// =====================================================================
__global__ __launch_bounds__(512)
void attn_topk_kernel(const float* __restrict__ qb, const float* __restrict__ kb,
                      const float* __restrict__ vb,
                      const float* __restrict__ w_attn, const float* __restrict__ b_attn,
                      const float* __restrict__ w_mil,  const float* __restrict__ b_mil,
                      float* __restrict__ probs, float* __restrict__ part)
{
    extern __shared__ float sm[];
    float* S  = sm;                          // [16][SSTR]
    float* TV = sm + 16 * SSTR;              // [16][KTOP] selected raw scores (desc)
    int*   TI = (int*)(TV + 16 * KTOP);      // [16][KTOP] selected key indices
    float* WL = (float*)(TI + 16 * KTOP);    // [16][2] per-row logits

    const int b    = blockIdx.x >> 8;
    const int qt   = blockIdx.x & 255;
    const int wave = threadIdx.x >> 5;
    const int lane = threadIdx.x & 31;
    const int rm   = lane & 15;
    const int kh   = lane >> 4;              // 0: K{0,1}, 1: K{2,3} of each 4-step
    const size_t bofs = (size_t)b * N_;

    // ---- A operand: Q tile rows, f32 16x4 layout, 3 K-steps (K = 0..11) ----
    const float* qrow = qb + (bofs + qt * 16 + rm) * QK_STR + kh * 2;
    v2f a0 = { qrow[0], qrow[1] };
    v2f a1 = { qrow[4], qrow[5] };
    v2f a2 = { qrow[8], qrow[9] };

    // ---- score stripe: each wave covers key tiles jt = wave, wave+16, ... ----
    for (int jt = wave; jt < 256; jt += 16) {
        const float* krow = kb + (bofs + jt * 16 + rm) * QK_STR + kh * 2;
        v2f b0 = { krow[0], krow[1] };
        v2f b1 = { krow[4], krow[5] };
        v2f b2 = { krow[8], krow[9] };
        v8f c = {};
        c = __builtin_amdgcn_wmma_f32_16x16x4_f32(false, a0, false, b0, (short)0, c, false, false);
        c = __builtin_amdgcn_wmma_f32_16x16x4_f32(false, a1, false, b1, (short)0, c, false, false);
        c = __builtin_amdgcn_wmma_f32_16x16x4_f32(false, a2, false, b2, (short)0, c, false, false);
        const int col = jt * 16 + rm;
        const int m0  = kh * 8;              // C/D layout: lanes 16-31 hold rows 8..15
        #pragma unroll
        for (int r = 0; r < 8; ++r) S[(m0 + r) * SSTR + col] = c[r];
    }
    __syncthreads();

    // ---- top-204 descending, one wave per query row ----
    const int row = wave;
    float* sr = S + row * SSTR;
    float lv = -INFINITY; int li = -1;       // cached max of this lane's strided strip
    for (int t = 0; t < 128; ++t) { int e = lane + (t << 5); float v = sr[e]; if (v > lv) { lv = v; li = e; } }

    for (int i = 0; i < KTOP; ++i) {
        float v = lv; int id = li;
        #pragma unroll
        for (int off = 16; off > 0; off >>= 1) {
            float ov = __shfl_xor(v, off, 32);
            int   oi = __shfl_xor(id, off, 32);
            if (ov > v || (ov == v && (unsigned)oi < (unsigned)id)) { v = ov; id = oi; }
        }
        if (lane == 0) { TV[row * KTOP + i] = v; TI[row * KTOP + i] = id; }
        if (lane == (id & 31)) {             // owner strip: remove and rescan
            sr[id] = -INFINITY;
            lv = -INFINITY; li = -1;
            for (int t = 0; t < 128; ++t) { int e = lane + (t << 5); float vv = sr[e]; if (vv > lv) { lv = vv; li = e; } }
        }
    }

    // ---- softmax over the 204 (scaled), probs out, ctx = probs @ gathered V ----
    const float m = TV[row * KTOP] * SCALE_;            // first selected == row max
    float ev[7];
    float esum = 0.0f;
    #pragma unroll
    for (int s2 = 0; s2 < 7; ++s2) {
        int gi = lane + (s2 << 5);
        float e = 0.0f;
        if (gi < KTOP) e = expf(TV[row * KTOP + gi] * SCALE_ - m);
        ev[s2] = e; esum += e;
    }
    #pragma unroll
    for (int off = 16; off > 0; off >>= 1) esum += __shfl_xor(esum, off, 32);
    const float inv = 1.0f / esum;

    float acc[HD_];
    #pragma unroll
    for (int d = 0; d < HD_; ++d) acc[d] = 0.0f;
    const size_t pbase = (bofs + qt * 16 + row) * (size_t)KTOP;
    #pragma unroll
    for (int s2 = 0; s2 < 7; ++s2) {
        int gi = lane + (s2 << 5);
        if (gi < KTOP) {
            float p = ev[s2] * inv;
            probs[pbase + gi] = p;
            const int id = TI[row * KTOP + gi];
            const float* vp = vb + (bofs + id) * QK_STR;
            #pragma unroll
            for (int d = 0; d < HD_; ++d) acc[d] = fmaf(p, vp[d], acc[d]);
        }
    }
    #pragma unroll
    for (int off = 16; off > 0; off >>= 1) {
        #pragma unroll
        for (int d = 0; d < HD_; ++d) acc[d] += __shfl_xor(acc[d], off, 32);
    }

    // ---- output projections (tiny): ctx -> 14 -> 2, per-row logits ----
    if (lane == 0) {
        float lg0 = b_mil[0], lg1 = b_mil[1];
        for (int o = 0; o < AFN_; ++o) {
            float t = b_attn[o];
            for (int d = 0; d < HD_; ++d) t = fmaf(w_attn[o * HD_ + d], acc[d], t);
            lg0 = fmaf(w_mil[o],        t, lg0);
            lg1 = fmaf(w_mil[AFN_ + o], t, lg1);
        }
        WL[wave * 2 + 0] = lg0; WL[wave * 2 + 1] = lg1;
    }
    __syncthreads();
    if (threadIdx.x < 2) {
        float s2 = 0.0f;
        for (int r = 0; r < 16; ++r) s2 += WL[r * 2 + threadIdx.x];
        part[(size_t)blockIdx.x * 2 + threadIdx.x] = s2;
    }
}

// =====================================================================
// Stage 3: deterministic reduction of per-tile logit partials -> pooled
// =====================================================================
__global__ void finalize_kernel(const float* __restrict__ part, float* __restrict__ out)
{
    const int t = threadIdx.x;
    if (t < B_ * 2) {
        const int b = t >> 1, c = t & 1;
        float s = 0.0f;
        for (int j = 0; j < 256; ++j) s += part[(size_t)(b * 256 + j) * 2 + c];
        out[t] = s * (1.0f / (float)N_);
    }
}

extern "C" void kernel_launch(void* const* d_in, const int* in_sizes, int n_in,
                              void* d_out, int out_size, void* d_ws, size_t ws_size,
                              hipStream_t stream)
{
    (void)in_sizes; (void)n_in; (void)out_size; (void)ws_size;
    const float* x    = (const float*)d_in[0];
    const float* cw2  = (const float*)d_in[1];  const float* cb2 = (const float*)d_in[2];
    const float* cw3  = (const float*)d_in[3];  const float* cb3 = (const float*)d_in[4];
    const float* cw4  = (const float*)d_in[5];  const float* cb4 = (const float*)d_in[6];
    const float* cw5  = (const float*)d_in[7];  const float* cb5 = (const float*)d_in[8];
    const float* cw6  = (const float*)d_in[9];  const float* cb6 = (const float*)d_in[10];
    const float* cw7  = (const float*)d_in[11]; const float* cb7 = (const float*)d_in[12];
    const float* w_q  = (const float*)d_in[13]; const float* b_q  = (const float*)d_in[14];
    const float* w_k  = (const float*)d_in[15]; const float* b_k  = (const float*)d_in[16];
    const float* w_v  = (const float*)d_in[17]; const float* b_v  = (const float*)d_in[18];
    const float* w_at = (const float*)d_in[19]; const float* b_at = (const float*)d_in[20];
    const float* w_mi = (const float*)d_in[21]; const float* b_mi = (const float*)d_in[22];

    // workspace layout (floats): q | k | v | per-tile logit partials
    float* ws   = (float*)d_ws;
    float* qbuf = ws;
    float* kbuf = ws + (size_t)B_ * N_ * QK_STR;         // 196608
    float* vbuf = ws + (size_t)2 * B_ * N_ * QK_STR;     // 393216
    float* part = ws + (size_t)3 * B_ * N_ * QK_STR;     // 589824 (+2048)

    float* out   = (float*)d_out;
    float* probs = out + B_ * 2;                         // pooled first, then probs

    feat_qkv_kernel<<<(B_ * N_) / 4, 128, 0, stream>>>(
        x, cw2, cb2, cw3, cb3, cw4, cb4, cw5, cb5, cw6, cb6, cw7, cb7,
        w_q, b_q, w_k, b_k, w_v, b_v, qbuf, kbuf, vbuf);

    const size_t shmem = (size_t)(16 * SSTR + 2 * 16 * KTOP + 32) * sizeof(float); // 288,640 B
    (void)hipFuncSetAttribute(reinterpret_cast<const void*>(attn_topk_kernel),
                              hipFuncAttributeMaxDynamicSharedMemorySize, (int)shmem);
    attn_topk_kernel<<<B_ * (N_ / 16), 512, shmem, stream>>>(
        qbuf, kbuf, vbuf, w_at, b_at, w_mi, b_mi, probs, part);

    finalize_kernel<<<1, 32, 0, stream>>>(part, out);
}